// NFP_53206054863434
// MI455X (gfx1250) — compile-verified
//
#include <hip/hip_runtime.h>

typedef float v2f __attribute__((ext_vector_type(2)));
typedef float v4f __attribute__((ext_vector_type(4)));
typedef float v8f __attribute__((ext_vector_type(8)));

#define NFP_M 10   // fingerprint dim (valid columns of every H)
#define RSTR  16   // row stride of nbr/rbuf: 64B rows, cols 10..15 = dump zone

// Single-instruction XOR lane shuffle: ds_swizzle_b32, group-of-32 mode,
// and_mask=0x1f, or_mask=0, xor_mask=MASK. EXEC is full at all uses.
template <int MASK>
__device__ __forceinline__ float swz_xor(float x) {
    return __int_as_float(
        __builtin_amdgcn_ds_swizzle(__float_as_int(x), (MASK << 10) | 0x1f));
}

// ---------------------------------------------------------------------------
// Prep (once, tiny): pre-pack the 4 H matrices pair-major, zero-padded to
// 12x16  (element (k,n) -> [(k/2)*32 + n*2 + (k&1)]), and zero f.
// A lane's B operand for K-pair kp at column n is then ONE float2 load.
// ---------------------------------------------------------------------------
__global__ __launch_bounds__(256)
void nfp_prep(const float* __restrict__ H0,
              const float* __restrict__ Hs,
              float* __restrict__ Hall,   // [4*192]
              float* __restrict__ f) {
    int i = blockIdx.x * blockDim.x + threadIdx.x;
    if (i < 4 * 192) {
        int hidx = i / 192, rem = i - hidx * 192;
        int kp = rem >> 5, t = rem & 31, n = t >> 1, d = t & 1;
        int k = 2 * kp + d;
        int in_dim = (hidx == 0) ? 6 : NFP_M;
        const float* src = (hidx == 0) ? H0 : (Hs + (hidx - 1) * NFP_M * NFP_M);
        Hall[i] = (k < in_dim && n < NFP_M) ? src[k * NFP_M + n] : 0.f;
    }
    if (i < 16) f[i] = 0.f;
}

// ---------------------------------------------------------------------------
__global__ __launch_bounds__(256)
void nfp_zero(float* __restrict__ nbr, int n) {
    int i = blockIdx.x * blockDim.x + threadIdx.x;
    if (i < n) nbr[i] = 0.0f;
}

// ---------------------------------------------------------------------------
// Edge scatter: nbr[dst] += xin[src]. FEAT=6 gathers x_member (stride-6 rows,
// 3x global_load_b64); FEAT=10 gathers rbuf (stride-16 rows, 64B aligned:
// b128+b128+b64). Scatter is FEAT fp32 atomics into the L2-resident nbr
// (192MB L2 vs ~80MB working set) -> bound by L2 RMW throughput.
// ---------------------------------------------------------------------------
template <int FEAT, int XSTRIDE>
__global__ __launch_bounds__(256)
void nfp_scatter(const float* __restrict__ xin,
                 const int* __restrict__ src, const int* __restrict__ dst,
                 float* __restrict__ nbr, int nE) {
    int e = blockIdx.x * blockDim.x + threadIdx.x;
    if (e >= nE) return;
    int s = src[e];
    int d = dst[e];
    const float* xs = xin + (size_t)s * XSTRIDE;
    float g[FEAT];
    if constexpr (FEAT == 6) {
        v2f p0 = *(const v2f*)(xs + 0);
        v2f p1 = *(const v2f*)(xs + 2);
        v2f p2 = *(const v2f*)(xs + 4);
        g[0] = p0.x; g[1] = p0.y; g[2] = p1.x;
        g[3] = p1.y; g[4] = p2.x; g[5] = p2.y;
    } else {
        v4f q0 = *(const v4f*)(xs + 0);
        v4f q1 = *(const v4f*)(xs + 4);
        v2f q2 = *(const v2f*)(xs + 8);
        g[0] = q0.x; g[1] = q0.y; g[2] = q0.z; g[3] = q0.w;
        g[4] = q1.x; g[5] = q1.y; g[6] = q1.z; g[7] = q1.w;
        g[8] = q2.x; g[9] = q2.y;
    }
    float* nd = nbr + (size_t)d * RSTR;
    #pragma unroll
    for (int j = 0; j < FEAT; ++j) atomicAdd(nd + j, g[j]);
}

// ---------------------------------------------------------------------------
// Node update with fp32 WMMA (V_WMMA_F32_16X16X4_F32):
//   h = xin[node] + nbr[node];  r = sigmoid(h @ H);
//   f += column-sums of softmax(r * w) over the 10 valid columns.
// Fully branchless:
//  - A-operand tail K-chunk: load offset clamped in-bounds, pair zero-masked
//    in registers (no reliance on zero padding in xin).
//  - r store: all 16 columns (cols 10-15 land in the row's dump zone);
//    out-of-range rows CLAMP to the dedicated dump row N (rbuf has N+1
//    rows) -> one v_min, no selects, no exec-mask branches.
//  - softmax: no max pass (inputs bounded by |w|, exp cannot overflow);
//    row sums are 4 batched steps of 8 independent ds_swizzles each.
// ---------------------------------------------------------------------------
template <int IN_DIM, int XSTRIDE>
__global__ __launch_bounds__(256)
void nfp_update(const float* __restrict__ xin,   // row stride XSTRIDE (may == rbuf)
                const float* __restrict__ nbr,   // [N*16]
                const float* __restrict__ Hp,    // [192] pair-major padded H
                const float* __restrict__ Wsc, int widx,
                float* __restrict__ rbuf,        // [(N+1)*16] out, row N = dump
                float* __restrict__ fglob,       // [16] accumulator
                int N) {
    constexpr int NC = (IN_DIM + 3) / 4;         // 2 for 6, 3 for 10
    const int lane = threadIdx.x & 31;
    const int wave = threadIdx.x >> 5;
    const int n    = lane & 15;                  // tile column
    const int hh   = lane >> 4;                  // half selector
    const int nodeBase = (blockIdx.x * (blockDim.x >> 5) + wave) * 16;

    // ---- load phase: all A/B operands up front, unconditional loads ----
    const int rowA = min(nodeBase + n, N - 1);   // clamped A row
    const float* xr = xin + (size_t)rowA * XSTRIDE;
    const float* nr = nbr + (size_t)rowA * RSTR;
    v2f av[NC], bv[NC];
    #pragma unroll
    for (int c = 0; c < NC; ++c) {
        const int k0 = 4 * c + 2 * hh;
        int k0c = k0;
        if (4 * c + 2 >= IN_DIM)                 // tail chunk only (folds)
            k0c = min(k0, IN_DIM - 2);           // keep load in-bounds
        v2f xv = *(const v2f*)(xr + k0c);        // 8B aligned (k0c even)
        v2f nv = *(const v2f*)(nr + k0c);
        v2f a  = xv + nv;
        if (4 * c + 2 >= IN_DIM) {               // zero-mask out-of-dim pair
            const bool kill = k0 >= IN_DIM;
            a.x = kill ? 0.f : a.x;
            a.y = kill ? 0.f : a.y;
        }
        av[c] = a;
        bv[c] = *(const v2f*)(Hp + ((2 * c + hh) * 16 + n) * 2);
    }

    // ---- back-to-back WMMAs ----
    v8f acc = {0.f, 0.f, 0.f, 0.f, 0.f, 0.f, 0.f, 0.f};
    #pragma unroll
    for (int c = 0; c < NC; ++c) {
        acc = __builtin_amdgcn_wmma_f32_16x16x4_f32(
            false, av[c], false, bv[c], (short)0, acc, false, false);
    }

    const float w = Wsc[widx];
    const bool ncol_ok = (n < NFP_M);
    bool  vrow[8];
    float rv[8];

    // ---- batch 1: sigmoids (8 independent exp/rcp chains pipeline) ----
    #pragma unroll
    for (int v = 0; v < 8; ++v) {
        rv[v]   = __builtin_amdgcn_rcpf(1.0f + __expf(-acc[v]));
        vrow[v] = (nodeBase + v + 8 * hh) < N;
    }

    // ---- batch 2: branchless stores (row clamped to dump row N) ----
    #pragma unroll
    for (int v = 0; v < 8; ++v) {
        const int snode = min(nodeBase + v + 8 * hh, N);
        rbuf[(size_t)snode * RSTR + n] = rv[v];
    }

    // ---- batch 3: softmax numerators (no max pass needed) ----
    float e[8], sum[8];
    #pragma unroll
    for (int v = 0; v < 8; ++v) {
        e[v] = (vrow[v] && ncol_ok) ? __expf(rv[v] * w) : 0.f;
        sum[v] = e[v];
    }

    // ---- batch 4: row sums; each step = 8 swizzles behind one wait ----
    float t[8];
    #pragma unroll
    for (int v = 0; v < 8; ++v) t[v] = swz_xor<8>(sum[v]);
    #pragma unroll
    for (int v = 0; v < 8; ++v) sum[v] += t[v];
    #pragma unroll
    for (int v = 0; v < 8; ++v) t[v] = swz_xor<4>(sum[v]);
    #pragma unroll
    for (int v = 0; v < 8; ++v) sum[v] += t[v];
    #pragma unroll
    for (int v = 0; v < 8; ++v) t[v] = swz_xor<2>(sum[v]);
    #pragma unroll
    for (int v = 0; v < 8; ++v) sum[v] += t[v];
    #pragma unroll
    for (int v = 0; v < 8; ++v) t[v] = swz_xor<1>(sum[v]);
    #pragma unroll
    for (int v = 0; v < 8; ++v) sum[v] += t[v];

    // ---- batch 5: probabilities -> per-column partial ----
    float colsum = 0.f;
    #pragma unroll
    for (int v = 0; v < 8; ++v) {
        colsum += (vrow[v] && ncol_ok)
                      ? e[v] * __builtin_amdgcn_rcpf(sum[v]) : 0.f;
    }
    colsum += swz_xor<16>(colsum);                // combine halves

    __shared__ float fblk[16];
    if (threadIdx.x < 16) fblk[threadIdx.x] = 0.f;
    __syncthreads();
    if (hh == 0 && ncol_ok) atomicAdd(&fblk[n], colsum);
    __syncthreads();
    if (threadIdx.x < NFP_M) atomicAdd(&fglob[threadIdx.x], fblk[threadIdx.x]);
}

// ---------------------------------------------------------------------------
// Finalize: group_out = x_group @ Wg; out = [f, group_out] @ Wm -> 3 floats.
// ---------------------------------------------------------------------------
__global__ void nfp_finalize(const float* __restrict__ f,
                             const float* __restrict__ x_group,
                             const float* __restrict__ Wg,
                             const float* __restrict__ Wm,
                             float* __restrict__ out) {
    int j = threadIdx.x;
    if (j >= 3) return;
    float go[4];
    #pragma unroll
    for (int q = 0; q < 4; ++q) {
        float a = 0.f;
        #pragma unroll
        for (int i = 0; i < 14; ++i) a += x_group[i] * Wg[i * 4 + q];
        go[q] = a;
    }
    float acc = 0.f;
    #pragma unroll
    for (int i = 0; i < 10; ++i) acc += f[i] * Wm[i * 3 + j];
    #pragma unroll
    for (int q = 0; q < 4; ++q) acc += go[q] * Wm[(10 + q) * 3 + j];
    out[j] = acc;
}

// ---------------------------------------------------------------------------
extern "C" void kernel_launch(void* const* d_in, const int* in_sizes, int n_in,
                              void* d_out, int out_size, void* d_ws, size_t ws_size,
                              hipStream_t stream) {
    const float* x_member = (const float*)d_in[0];  // [N,6]
    const float* x_group  = (const float*)d_in[1];  // [1,14]
    const int*   esrc     = (const int*)d_in[2];    // [nE]
    const int*   edst     = (const int*)d_in[3];    // [nE]
    const float* H0       = (const float*)d_in[4];  // [6,10]
    const float* Hs       = (const float*)d_in[5];  // [3,10,10]
    const float* Wsc      = (const float*)d_in[6];  // [4]
    const float* Wg       = (const float*)d_in[7];  // [14,4]
    const float* Wm       = (const float*)d_in[8];  // [14,3]

    const int N  = in_sizes[0] / 6;
    const int nE = in_sizes[2];

    float* nbr  = (float*)d_ws;                      // [N*16]
    float* rbuf = nbr  + (size_t)N * RSTR;           // [(N+1)*16], row N = dump
    float* Hall = rbuf + (size_t)(N + 1) * RSTR;     // [4*192]
    float* f    = Hall + 4 * 192;                    // [16]

    const dim3 blk(256);
    const int zg = (N * RSTR + 255) / 256;
    const int sg = (nE + 255) / 256;
    const int ug = (N + 127) / 128;                  // 8 waves/block, 16 nodes/wave

    nfp_prep<<<3, blk, 0, stream>>>(H0, Hs, Hall, f);

    // Round 0: t=6 neighbor sum, project through H0 (2 WMMAs per tile)
    nfp_zero<<<zg, blk, 0, stream>>>(nbr, N * RSTR);
    nfp_scatter<6, 6><<<sg, blk, 0, stream>>>(x_member, esrc, edst, nbr, nE);
    nfp_update<6, 6><<<ug, blk, 0, stream>>>(x_member, nbr, Hall, Wsc, 0,
                                             rbuf, f, N);

    // Rounds 1..3: m=10 space, H = Hs[L-1] (3 WMMAs per tile)
    for (int L = 1; L <= 3; ++L) {
        nfp_zero<<<zg, blk, 0, stream>>>(nbr, N * RSTR);
        nfp_scatter<10, RSTR><<<sg, blk, 0, stream>>>(rbuf, esrc, edst, nbr, nE);
        nfp_update<10, RSTR><<<ug, blk, 0, stream>>>(rbuf, nbr,
                                                     Hall + (size_t)L * 192,
                                                     Wsc, L, rbuf, f, N);
    }

    nfp_finalize<<<1, 32, 0, stream>>>(f, x_group, Wg, Wm, (float*)d_out);
}